// NBC_28716151341229
// MI455X (gfx1250) — compile-verified
//
#include <hip/hip_runtime.h>
#include <math.h>

typedef __attribute__((ext_vector_type(2))) float v2f;
typedef __attribute__((ext_vector_type(8))) float v8f;

#define C_CLASSES 32
#define F_CAT     16
#define F_NUM     16
#define VOCAB     64
#define LCAT_ELEMS (F_CAT * VOCAB * C_CLASSES)          /* 32768 floats = 128 KB */
#define SMEM_FLOATS (LCAT_ELEMS + 512 + 512 + 32)       /* + coefA + coefB + const */
#define WAVES_PER_BLOCK 8

// Order-preserving map f32 -> u32 (total order; -0 < +0 irrelevant for argmax).
__device__ __forceinline__ unsigned int fkey(float f) {
    unsigned int b = __float_as_uint(f);
    return b ^ (unsigned int)(((int)b >> 31) | 0x80000000);
}

__global__ __launch_bounds__(256) void nbc_wmma_kernel(
    const int*   __restrict__ Xcat,        // (N,16) int32
    const float* __restrict__ Xnum,        // (N,16) f32
    const float* __restrict__ class_probs, // (32,)
    const float* __restrict__ cat_probs,   // (32,16,64)
    const float* __restrict__ means,       // (32,16)
    const float* __restrict__ stds,        // (32,16)
    float*       __restrict__ out,         // (N,) argmax index (as float)
    int n)
{
    extern __shared__ float smem[];
    float* s_lcat = smem;                 // [f][v][c] log cat_probs, c contiguous
    float* s_cA   = smem + LCAT_ELEMS;    // [f][c]  -0.5/sigma^2
    float* s_cB   = s_cA + 512;           // [f][c]  mean/sigma^2
    float* s_cst  = s_cB + 512;           // [c]     log P(c) + quad/normalizer consts

    const int tid = threadIdx.x;

    // ---- per-block table build (tables tiny; source reads L2-resident) ----
    for (int i = tid; i < LCAT_ELEMS; i += 256) {
        int c = i & 31, v = (i >> 5) & 63, f = i >> 11;
        s_lcat[i] = __logf(cat_probs[(c * F_CAT + f) * VOCAB + v]);
    }
    for (int i = tid; i < 512; i += 256) {
        int c = i & 31, f = i >> 5;
        float s   = stds[c * F_NUM + f];
        float inv = 1.0f / (s * s);
        s_cA[i] = -0.5f * inv;
        s_cB[i] = means[c * F_NUM + f] * inv;
    }
    if (tid < C_CLASSES) {
        const float LOG2PI = 1.8378770664093453f;
        float acc = __logf(class_probs[tid]);
        #pragma unroll
        for (int f = 0; f < F_NUM; ++f) {
            float s = stds[tid * F_NUM + f];
            float m = means[tid * F_NUM + f];
            acc += -0.5f * m * m / (s * s) - LOG2PI - 2.0f * __logf(s);
        }
        s_cst[tid] = acc;
    }
    __syncthreads();

    const int lane = tid & 31;
    const int wave = tid >> 5;
    const int half = lane >> 4;   // half-wave: K sub-slot / sample-row offset
    const int l16  = lane & 15;

    // ---- B fragments: constant across all sample tiles, load once per wave ----
    // V_WMMA_F32_16X16X4_F32 B layout: VGPR0 = row K=4kb+2*half, VGPR1 = K=4kb+2*half+1,
    // column N = lane%16 (+16*t for the second class tile).
    v2f   bfrag[2][8];
    float cst[2];
    #pragma unroll
    for (int t = 0; t < 2; ++t) {
        int c = l16 + 16 * t;
        cst[t] = s_cst[c];
        #pragma unroll
        for (int kb = 0; kb < 8; ++kb) {
            int f = 2 * kb + half;            // K index pair (2f, 2f+1)
            v2f b;
            b[0] = s_cA[f * 32 + c];
            b[1] = s_cB[f * 32 + c];
            bfrag[t][kb] = b;
        }
    }

    const int ntiles = (n + 15) >> 4;
    for (int tile = blockIdx.x * WAVES_PER_BLOCK + wave; tile < ntiles;
         tile += gridDim.x * WAVES_PER_BLOCK) {
        const int s0 = tile << 4;

        // ---- A fragments: A[m][2f]=x^2, A[m][2f+1]=x ; row m = lane%16 ----
        int arow = s0 + l16; if (arow >= n) arow = n - 1;
        const float* xrow = Xnum + (size_t)arow * F_NUM;
        v2f afrag[8];
        #pragma unroll
        for (int kb = 0; kb < 8; ++kb) {
            float x = xrow[2 * kb + half];    // feature f = 2kb + half
            v2f a; a[0] = x * x; a[1] = x;
            afrag[kb] = a;
        }

        // ---- C init: log prior + Gaussian consts + categorical gather-sum (LDS) ----
        // C layout: VGPR j <-> sample row j + 8*half ; lane%16 <-> class column.
        v8f c0, c1;
        #pragma unroll
        for (int j = 0; j < 8; ++j) {
            int samp = s0 + j + 8 * half; if (samp >= n) samp = n - 1;
            const int4* cr = (const int4*)(Xcat + (size_t)samp * F_CAT);
            int4 w0 = cr[0], w1 = cr[1], w2 = cr[2], w3 = cr[3];
            int idx[16] = { w0.x, w0.y, w0.z, w0.w,  w1.x, w1.y, w1.z, w1.w,
                            w2.x, w2.y, w2.z, w2.w,  w3.x, w3.y, w3.z, w3.w };
            float a0 = cst[0], a1 = cst[1];
            #pragma unroll
            for (int f = 0; f < F_CAT; ++f) {
                const float* p = s_lcat + (((f << 6) + idx[f]) << 5) + l16;
                a0 += p[0];       // classes 0..15
                a1 += p[16];      // classes 16..31
            }
            c0[j] = a0;
            c1[j] = a1;
        }

        // ---- Gaussian quadratic via fp32 WMMA: 8 K-steps x 2 class tiles ----
        #pragma unroll
        for (int kb = 0; kb < 8; ++kb) {
            c0 = __builtin_amdgcn_wmma_f32_16x16x4_f32(
                false, afrag[kb], false, bfrag[0][kb], (short)0, c0, false, false);
            c1 = __builtin_amdgcn_wmma_f32_16x16x4_f32(
                false, afrag[kb], false, bfrag[1][kb], (short)0, c1, false, false);
        }

        // ---- branchless argmax over 32 classes ----
        // Pack (monotonic score key << 6) | (63 - class): plain u64 max reduces,
        // ties pick the smaller class index (first-match, like jnp.argmax).
        #pragma unroll
        for (int j = 0; j < 8; ++j) {
            unsigned long long k0 =
                ((unsigned long long)fkey(c0[j]) << 6) | (unsigned)(63 - l16);
            unsigned long long k1 =
                ((unsigned long long)fkey(c1[j]) << 6) | (unsigned)(63 - (l16 + 16));
            unsigned long long best = (k1 > k0) ? k1 : k0;
            #pragma unroll
            for (int mask = 1; mask < 16; mask <<= 1) {
                unsigned long long o = __shfl_xor(best, mask, 32);
                best = (o > best) ? o : best;
            }
            int samp = s0 + j + 8 * half;
            if (l16 == 0 && samp < n)
                out[samp] = (float)(63 - (int)(best & 63ull));
        }

        // ---- prefetch next tile's rows (wave-uniform branch) ----
        int nt = tile + gridDim.x * WAVES_PER_BLOCK;
        if (nt < ntiles) {
            int pr = (nt << 4) + l16; if (pr >= n) pr = n - 1;
            __builtin_prefetch(Xnum + (size_t)pr * F_NUM, 0, 1);
            __builtin_prefetch(Xcat + (size_t)pr * F_CAT, 0, 1);
        }
    }
}

extern "C" void kernel_launch(void* const* d_in, const int* in_sizes, int n_in,
                              void* d_out, int out_size, void* d_ws, size_t ws_size,
                              hipStream_t stream) {
    const int*   Xcat = (const int*)d_in[0];
    const float* Xnum = (const float*)d_in[1];
    const float* cp   = (const float*)d_in[2];
    const float* catp = (const float*)d_in[3];
    const float* mn   = (const float*)d_in[4];
    const float* sd   = (const float*)d_in[5];
    float* out = (float*)d_out;

    const int n = in_sizes[0] / F_CAT;          // N samples
    const size_t smem_bytes = (size_t)SMEM_FLOATS * sizeof(float);  // ~135 KB
    (void)hipFuncSetAttribute(reinterpret_cast<const void*>(nbc_wmma_kernel),
                              hipFuncAttributeMaxDynamicSharedMemorySize,
                              (int)smem_bytes);

    const int ntiles = (n + 15) >> 4;
    int blocks = 512;                            // ~2 blocks/WGP, 135KB LDS each
    int maxb = (ntiles + WAVES_PER_BLOCK - 1) / WAVES_PER_BLOCK;
    if (blocks > maxb) blocks = maxb;
    if (blocks < 1) blocks = 1;

    nbc_wmma_kernel<<<blocks, 256, smem_bytes, stream>>>(
        Xcat, Xnum, cp, catp, mn, sd, out, n);
}